// RoutingCapsules_75995151335439
// MI455X (gfx1250) — compile-verified
//
#include <hip/hip_runtime.h>

// Problem constants (from reference): x (16,2048,64) f32, W (64,32,64) f32,
// out (16,64,32) f32.
#define BB 16
#define II 2048
#define KK 64
#define JJ 64
#define DD 32

typedef __attribute__((ext_vector_type(2))) float v2f;
typedef __attribute__((ext_vector_type(8))) float v8f;

// ---------------------------------------------------------------------------
// xsum[b,k] = sum_i x[b,i,k]
// ---------------------------------------------------------------------------
__global__ void xsum_kernel(const float* __restrict__ x, float* __restrict__ xsum) {
  int b = blockIdx.x;
  int tid = threadIdx.x;            // 256
  int k = tid & 63, g = tid >> 6;   // g in 0..3
  const float* xb = x + (size_t)b * II * KK;
  float p = 0.f;
  for (int i = g; i < II; i += 4) p += xb[i * KK + k];
  __shared__ float red[256];
  red[tid] = p;
  __syncthreads();
  if (tid < 64)
    xsum[b * KK + tid] = red[tid] + red[tid + 64] + red[tid + 128] + red[tid + 192];
}

// ---------------------------------------------------------------------------
// s[b,j,d] = scale * sum_k W[j,d,k] * cx[b,(j),k] ; v = squash(s)
// optional: WvT[b,k,j] = sum_d W[j,d,k] * v[b,j,d] ; optional: out[b,j,d] = v
// blockDim = 64 (wave0 = d lanes), grid = (J, B)
// ---------------------------------------------------------------------------
__global__ void sv_kernel(const float* __restrict__ W, const float* __restrict__ cx,
                          int cx_bstride, int cx_jstride, float scale,
                          float* __restrict__ WvT, float* __restrict__ outv) {
  int j = blockIdx.x, b = blockIdx.y;
  int tid = threadIdx.x;
  const float* wj = W + (size_t)j * DD * KK;
  const float* cxv = cx + (size_t)b * cx_bstride + (size_t)j * cx_jstride;
  __shared__ float v_s[DD];
  if (tid < DD) {
    float s = 0.f;
    for (int k = 0; k < KK; ++k) s += wj[tid * KK + k] * cxv[k];
    s *= scale;
    float sq = s * s;  // wave32 reduction over the 32 d-lanes
    for (int off = 16; off > 0; off >>= 1) sq += __shfl_xor(sq, off, 32);
    float v = s * (sq / ((1.f + sq) * (sqrtf(sq) + 1e-8f)));
    v_s[tid] = v;
    if (outv) outv[(size_t)b * JJ * DD + (size_t)j * DD + tid] = v;
  }
  __syncthreads();
  if (WvT) {
    int k = tid;  // 0..63
    float wv = 0.f;
    for (int d = 0; d < DD; ++d) wv += wj[d * KK + k] * v_s[d];
    WvT[(size_t)b * KK * JJ + (size_t)k * JJ + j] = wv;
  }
}

// ---------------------------------------------------------------------------
// bl[b,i,j] (+)= sum_k x[b,i,k] * WvT[b,k,j]   (per-batch 2048x64 @ 64x64)
// V_WMMA_F32_16X16X4_F32; block = 128 thr = 4 waves, one i-tile, 4 j-tiles.
// ---------------------------------------------------------------------------
__global__ void blogit_kernel(const float* __restrict__ x, const float* __restrict__ WvT,
                              float* __restrict__ bl, int accumulate) {
  int bid = blockIdx.x;                 // B * 128 i-tiles
  int b = bid >> 7, it = bid & 127;
  int tid = threadIdx.x;                // 128
  int lane = tid & 31, wave = tid >> 5;
  int half = lane >> 4, l16 = lane & 15;
  __shared__ float xs[16 * 64];         // x tile, row-major
  __shared__ float wvs[64 * 64];        // WvT[b] (k-major)
  const float* xb = x + (size_t)b * II * KK + (size_t)it * 16 * KK;
  for (int idx = tid; idx < 16 * 64; idx += 128) xs[idx] = xb[idx];
  const float* wvb = WvT + (size_t)b * KK * JJ;
  for (int idx = tid; idx < 64 * 64; idx += 128) wvs[idx] = wvb[idx];
  __syncthreads();

  int j0 = wave * 16;
  float* blb = bl + (size_t)b * II * JJ + (size_t)it * 16 * JJ;
  v8f acc;
  if (accumulate) {
#pragma unroll
    for (int r = 0; r < 8; ++r) acc[r] = blb[(r + 8 * half) * JJ + j0 + l16];
  } else {
#pragma unroll
    for (int r = 0; r < 8; ++r) acc[r] = 0.f;
  }
#pragma unroll
  for (int kk = 0; kk < 16; ++kk) {     // K = 64, 4 per WMMA
    int k0 = kk * 4 + 2 * half;
    v2f a, bf;
    a[0] = xs[l16 * 64 + k0];           // A[M=i][K]: lanes<16 K=0,1; lanes>=16 K=2,3
    a[1] = xs[l16 * 64 + k0 + 1];
    bf[0] = wvs[k0 * 64 + j0 + l16];    // B[K][N=j]
    bf[1] = wvs[(k0 + 1) * 64 + j0 + l16];
    acc = __builtin_amdgcn_wmma_f32_16x16x4_f32(false, a, false, bf, (short)0, acc,
                                                false, false);
  }
#pragma unroll
  for (int r = 0; r < 8; ++r) blb[(r + 8 * half) * JJ + j0 + l16] = acc[r];
}

// ---------------------------------------------------------------------------
// c[b,i,j] = softmax_j(bl[b,i,j]) ; 64 contiguous logits per row
// ---------------------------------------------------------------------------
__global__ void softmax_kernel(const float* __restrict__ bl, float* __restrict__ c) {
  int tid = threadIdx.x;              // 256 = 4 rows x 64
  int r = tid >> 6, t = tid & 63;
  size_t row = (size_t)blockIdx.x * 4 + r;
  float val = bl[row * 64 + t];
  __shared__ float red[256];
  red[tid] = val;
  __syncthreads();
  for (int off = 32; off > 0; off >>= 1) {
    if (t < off) red[tid] = fmaxf(red[tid], red[tid + off]);
    __syncthreads();
  }
  float m = red[r * 64];
  __syncthreads();
  float e = expf(val - m);
  red[tid] = e;
  __syncthreads();
  for (int off = 32; off > 0; off >>= 1) {
    if (t < off) red[tid] += red[tid + off];
    __syncthreads();
  }
  float s = red[r * 64];
  c[row * 64 + t] = e / s;
}

__global__ void zero_kernel(float* __restrict__ p, int n) {
  int i = blockIdx.x * blockDim.x + threadIdx.x;
  if (i < n) p[i] = 0.f;
}

// ---------------------------------------------------------------------------
// cx[b,j,k] = sum_i c[b,i,j] * x[b,i,k]  (per-batch 64x2048 @ 2048x64)
// grid (8 i-splits, B); block 512 = 16 waves, each wave one 16x16 (j,k) tile.
// Reduction over i via WMMA K-dim; i-splits combined with f32 global atomics.
// ---------------------------------------------------------------------------
__global__ void cx_kernel(const float* __restrict__ c, const float* __restrict__ x,
                          float* __restrict__ cx) {
  int isplit = blockIdx.x, b = blockIdx.y;
  int tid = threadIdx.x;                 // 512
  int lane = tid & 31, wave = tid >> 5;  // 16 waves
  int half = lane >> 4, l16 = lane & 15;
  int jt = wave >> 2, kt = wave & 3;
  int j0 = jt * 16, k0c = kt * 16;
  __shared__ float cs[32 * 64];
  __shared__ float xs[32 * 64];
  v8f acc;
#pragma unroll
  for (int r = 0; r < 8; ++r) acc[r] = 0.f;
  const float* cb = c + (size_t)b * II * JJ;
  const float* xb = x + (size_t)b * II * KK;
  for (int chunk = 0; chunk < 8; ++chunk) {  // 256 i's per split, 32 per chunk
    int ibase = isplit * 256 + chunk * 32;
    for (int idx = tid; idx < 2048; idx += 512) {
      cs[idx] = cb[(size_t)ibase * 64 + idx];
      xs[idx] = xb[(size_t)ibase * 64 + idx];
    }
    __syncthreads();
#pragma unroll
    for (int s = 0; s < 8; ++s) {
      int i0 = s * 4 + 2 * half;
      v2f a, bf;
      a[0] = cs[i0 * 64 + j0 + l16];        // A[M=j][K=i] = c[i][j]
      a[1] = cs[(i0 + 1) * 64 + j0 + l16];
      bf[0] = xs[i0 * 64 + k0c + l16];      // B[K=i][N=k] = x[i][k]
      bf[1] = xs[(i0 + 1) * 64 + k0c + l16];
      acc = __builtin_amdgcn_wmma_f32_16x16x4_f32(false, a, false, bf, (short)0, acc,
                                                  false, false);
    }
    __syncthreads();
  }
#pragma unroll
  for (int r = 0; r < 8; ++r)
    atomicAdd(&cx[(size_t)b * JJ * KK + (size_t)(j0 + r + 8 * half) * KK + k0c + l16],
              acc[r]);
}

// ---------------------------------------------------------------------------
extern "C" void kernel_launch(void* const* d_in, const int* in_sizes, int n_in,
                              void* d_out, int out_size, void* d_ws, size_t ws_size,
                              hipStream_t stream) {
  (void)in_sizes; (void)n_in; (void)out_size; (void)ws_size;
  const float* x = (const float*)d_in[0];   // (16,2048,64)
  const float* W = (const float*)d_in[1];   // (64,32,64)
  float* out = (float*)d_out;               // (16,64,32)

  float* ws   = (float*)d_ws;
  float* bl   = ws;                               // B*I*J  = 2,097,152 f
  float* c    = bl + (size_t)BB * II * JJ;        // B*I*J  = 2,097,152 f
  float* xsum = c + (size_t)BB * II * JJ;         // B*K    = 1,024 f
  float* cxb  = xsum + BB * KK;                   // B*J*K  = 65,536 f
  float* WvT  = cxb + BB * JJ * KK;               // B*K*J  = 65,536 f

  // Round 1: uniform coupling (softmax of zeros) -> s1 = (1/64) W @ xsum
  xsum_kernel<<<BB, 256, 0, stream>>>(x, xsum);
  sv_kernel<<<dim3(JJ, BB), 64, 0, stream>>>(W, xsum, KK, 0, 1.0f / 64.0f, WvT, nullptr);
  blogit_kernel<<<BB * (II / 16), 128, 0, stream>>>(x, WvT, bl, 0);  // b-update #1

  for (int round = 1; round < 3; ++round) {
    softmax_kernel<<<(BB * II) / 4, 256, 0, stream>>>(bl, c);
    zero_kernel<<<(BB * JJ * KK) / 256, 256, 0, stream>>>(cxb, BB * JJ * KK);
    cx_kernel<<<dim3(8, BB), 512, 0, stream>>>(c, x, cxb);
    if (round < 2) {
      sv_kernel<<<dim3(JJ, BB), 64, 0, stream>>>(W, cxb, JJ * KK, KK, 1.0f, WvT, nullptr);
      blogit_kernel<<<BB * (II / 16), 128, 0, stream>>>(x, WvT, bl, 1);  // b-update #2
    } else {
      // Final: out = squash(W @ (softmax(b) @ x)) per (b,j)
      sv_kernel<<<dim3(JJ, BB), 64, 0, stream>>>(W, cxb, JJ * KK, KK, 1.0f, nullptr, out);
    }
  }
}